// GNCC_49306224558823
// MI455X (gfx1250) — compile-verified
//
#include <hip/hip_runtime.h>
#include <math.h>

typedef float v2f __attribute__((ext_vector_type(2)));
typedef float v8f __attribute__((ext_vector_type(8)));

#define N_NODES 8192
#define N_EDGES 65536
#define N_ITEMS (N_EDGES * 8)      // edge-corner pairs
#define NBINS   15625              // KTOT = 25^3 weight matrices
#define PAD     32
#define EPS_BN  1e-5f
#define TILE_M  16
#define MAXTILES 48400             // >= sum ceil(cnt_k/16) <= 32768 + 15625
#define MAXPAD   (MAXTILES * TILE_M)
#define SCAN_BLOCKS 62             // 62*256 = 15872 >= NBINS

__device__ __forceinline__ int imin(int a, int b) { return a < b ? a : b; }

// ---------------------------------------------------------------------------
// One-time: per-edge 8 corner indices + trilinear basis, degree, bin histogram
// ---------------------------------------------------------------------------
__global__ __launch_bounds__(256) void precompute_kernel(
    const float* __restrict__ pseudo, const int* __restrict__ dst,
    float* __restrict__ ebas, int* __restrict__ ecorner,
    float* __restrict__ deg, int* __restrict__ cnt)
{
    int e = blockIdx.x * blockDim.x + threadIdx.x;
    if (e >= N_EDGES) return;
    float fr[3]; int lo[3];
#pragma unroll
    for (int d = 0; d < 3; d++) {
        float v = pseudo[e * 3 + d] * 24.0f;           // (K1-1)
        float lf = floorf(v);
        lf = fminf(fmaxf(lf, 0.0f), 23.0f);            // clip to [0, K1-2]
        fr[d] = v - lf;
        lo[d] = (int)lf;
    }
#pragma unroll
    for (int s = 0; s < 8; s++) {
        float basis = 1.0f; int idx = 0; int stride = 1;
#pragma unroll
        for (int d = 0; d < 3; d++) {
            int bit = (s >> d) & 1;
            basis *= bit ? fr[d] : (1.0f - fr[d]);
            idx += (lo[d] + bit) * stride;
            stride *= 25;
        }
        ebas[e * 8 + s]    = basis;
        ecorner[e * 8 + s] = idx;
        atomicAdd(&cnt[idx], 1);
    }
    atomicAdd(&deg[dst[e]], 1.0f);
}

__global__ __launch_bounds__(256) void invdeg_kernel(
    const float* __restrict__ deg, float* __restrict__ invdeg)
{
    int n = blockIdx.x * blockDim.x + threadIdx.x;
    if (n < N_NODES) invdeg[n] = 1.0f / fmaxf(deg[n], 1.0f);
}

__global__ __launch_bounds__(256) void init_h_kernel(
    const float* __restrict__ x, float* __restrict__ h)
{
    int i = blockIdx.x * blockDim.x + threadIdx.x;
    if (i < N_NODES * PAD) h[i] = ((i & (PAD - 1)) == 0) ? x[i >> 5] : 0.0f;
}

// ---------------------------------------------------------------------------
// Counting-sort scaffolding: block-local scans of (tile count, padded count)
// ---------------------------------------------------------------------------
__global__ __launch_bounds__(256) void scan_local_kernel(
    const int* __restrict__ cnt, int* __restrict__ tileOffL,
    int* __restrict__ padOffL, int* __restrict__ blkNT, int* __restrict__ blkPC)
{
    __shared__ int sNT[256], sPC[256];
    int t = threadIdx.x;
    int bin = blockIdx.x * 256 + t;
    int c  = (bin < NBINS) ? cnt[bin] : 0;
    int nt = (c + 15) >> 4;
    int pc = nt << 4;
    sNT[t] = nt; sPC[t] = pc;
    __syncthreads();
    for (int off = 1; off < 256; off <<= 1) {      // Hillis-Steele inclusive scan
        int vNT = 0, vPC = 0;
        if (t >= off) { vNT = sNT[t - off]; vPC = sPC[t - off]; }
        __syncthreads();
        sNT[t] += vNT; sPC[t] += vPC;
        __syncthreads();
    }
    if (bin < NBINS) {
        tileOffL[bin] = sNT[t] - nt;               // block-local exclusive
        padOffL[bin]  = sPC[t] - pc;
    }
    if (t == 255) { blkNT[blockIdx.x] = sNT[255]; blkPC[blockIdx.x] = sPC[255]; }
}

__global__ void scan_blocks_kernel(int* blkNT, int* blkPC, int* nTiles)
{
    if (blockIdx.x == 0 && threadIdx.x == 0) {     // tiny serial scan over 62
        int aNT = 0, aPC = 0;
        for (int i = 0; i < SCAN_BLOCKS; i++) {
            int nt = blkNT[i], pc = blkPC[i];
            blkNT[i] = aNT; blkPC[i] = aPC;
            aNT += nt; aPC += pc;
        }
        nTiles[0] = aNT;
    }
}

__global__ __launch_bounds__(256) void build_tiles_kernel(
    const int* __restrict__ cnt, int* __restrict__ tileOffL,
    int* __restrict__ padOffL, const int* __restrict__ blkNT,
    const int* __restrict__ blkPC, int* __restrict__ tileK,
    int* __restrict__ tileBase)
{
    int bin = blockIdx.x * 256 + threadIdx.x;
    if (bin >= NBINS) return;
    int to = tileOffL[bin] + blkNT[bin >> 8];
    int po = padOffL[bin]  + blkPC[bin >> 8];
    tileOffL[bin] = to;                            // global tile offset
    padOffL[bin]  = po;                            // global padded row offset
    int nt = (cnt[bin] + 15) >> 4;
    for (int t = 0; t < nt; t++) { tileK[to + t] = bin; tileBase[to + t] = po + t * TILE_M; }
}

__global__ __launch_bounds__(256) void scatter_kernel(
    const int* __restrict__ ecorner, const float* __restrict__ ebas,
    const int* __restrict__ padOff, int* __restrict__ fill,
    int* __restrict__ sortedE, float* __restrict__ sortedBas)
{
    int i = blockIdx.x * blockDim.x + threadIdx.x;
    if (i >= N_ITEMS) return;
    int k = ecorner[i];
    int pos = padOff[k] + atomicAdd(&fill[k], 1);
    sortedE[pos]   = i >> 3;
    sortedBas[pos] = ebas[i];
}

// ---------------------------------------------------------------------------
// Edge pass as grouped GEMM: one wave = one 16-row tile sharing W[k].
// Compile-time CI/CO: guards fold away for 32x32 layers; CO<=16 uses a single
// N-tile; K-loop runs ceil(CI/4) steps. B indices are clamped (branchless,
// memory-safe); A rows are zero beyond CI (padded h), so clamped B values
// contribute exact zeros. D scaled per-row by basis, scattered via f32 atomics.
// ---------------------------------------------------------------------------
template <int CI, int CO>
__global__ __launch_bounds__(256) void edge_tile_kernel(
    const float* __restrict__ hIn, const int* __restrict__ src,
    const int* __restrict__ dst, const int* __restrict__ sortedE,
    const float* __restrict__ sortedBas, const int* __restrict__ tileK,
    const int* __restrict__ tileBase, const int* __restrict__ nTiles,
    const float* __restrict__ W, float* __restrict__ agg)
{
    constexpr int  KB  = (CI + 3) & ~3;            // K extent rounded to 4
    constexpr bool TWO = (CO > 16);                // second N-tile exists

    int wave = (blockIdx.x * blockDim.x + threadIdx.x) >> 5;
    if (wave >= nTiles[0]) return;                 // wave-uniform exit
    int lane  = threadIdx.x & 31;
    int n15   = lane & 15;
    int khalf = lane >> 4;

    int k    = tileK[wave];
    int base = tileBase[wave];
    const float* Wk = W + (size_t)k * (CI * CO);
    __builtin_prefetch(Wk, 0, 3);                  // global_prefetch_b8: pull W[k]

    int eRow = sortedE[base + n15];                // edge feeding my A-matrix row
    const float* arow = hIn + (size_t)src[eRow] * PAD;

    int o0 = imin(n15, CO - 1);                    // branchless clamp (folds if CO>=16)

    v8f acc0 = {0.f,0.f,0.f,0.f,0.f,0.f,0.f,0.f};
    v8f acc1 = {0.f,0.f,0.f,0.f,0.f,0.f,0.f,0.f};
#pragma unroll
    for (int kb = 0; kb < KB; kb += 4) {
        int k0 = kb + 2 * khalf;
        int i0 = imin(k0,     CI - 1);             // folds away when CI==32
        int i1 = imin(k0 + 1, CI - 1);
        v2f a;  a.x  = arow[k0];           a.y  = arow[k0 + 1];
        v2f b0; b0.x = Wk[i0 * CO + o0];   b0.y = Wk[i1 * CO + o0];
        acc0 = __builtin_amdgcn_wmma_f32_16x16x4_f32(false, a, false, b0,
                                                     (short)0, acc0, false, false);
        if constexpr (TWO) {
            v2f b1; b1.x = Wk[i0 * CO + n15 + 16]; b1.y = Wk[i1 * CO + n15 + 16];
            acc1 = __builtin_amdgcn_wmma_f32_16x16x4_f32(false, a, false, b1,
                                                         (short)0, acc1, false, false);
        }
    }

    // D fragment row r -> padded row (r + 8*khalf); scale by basis, scatter.
#pragma unroll
    for (int r = 0; r < 8; r++) {
        int row   = base + r + 8 * khalf;
        float bas = sortedBas[row];                // 0 for padding rows
        int e     = sortedE[row];
        int dN    = dst[e];
        if (bas != 0.0f) {
            if (n15 < CO) atomicAdd(&agg[dN * PAD + n15], acc0[r] * bas);
            if constexpr (TWO)
                atomicAdd(&agg[dN * PAD + n15 + 16], acc1[r] * bas);
        }
    }
}

// ---------------------------------------------------------------------------
// Fused node kernel: D = hIn @ rootPad via V_WMMA_F32_16X16X4_F32, then
// + agg*invdeg + bias, ELU, per-channel sum/sumsq atomics (BN stats).
// ---------------------------------------------------------------------------
__global__ __launch_bounds__(256) void node_gemm_kernel(
    const float* __restrict__ hIn, const float* __restrict__ agg,
    const float* __restrict__ invdeg, const float* __restrict__ root,
    const float* __restrict__ bias, float* __restrict__ hAct,
    float* __restrict__ stats, float* __restrict__ finalOut,
    int ci, int co, int isFinal)
{
    __shared__ float rootp[32 * 32];               // zero-padded root (K x N)
    int t = threadIdx.x;
    for (int i = t; i < 1024; i += 256) {
        int k = i >> 5, n = i & 31;
        rootp[i] = (k < ci && n < co) ? root[k * co + n] : 0.0f;
    }
    __syncthreads();

    int wave = t >> 5, lane = t & 31;
    int rowBase = blockIdx.x * 128 + wave * 16;
    int n15   = lane & 15;
    int khalf = lane >> 4;

    v8f acc0 = {0.f,0.f,0.f,0.f,0.f,0.f,0.f,0.f};
    v8f acc1 = {0.f,0.f,0.f,0.f,0.f,0.f,0.f,0.f};
    const float* arow = hIn + (size_t)(rowBase + n15) * PAD;

#pragma unroll
    for (int kb = 0; kb < 32; kb += 4) {
        int k0 = kb + 2 * khalf;
        v2f a;  a.x  = arow[k0];                  a.y  = arow[k0 + 1];
        v2f b0; b0.x = rootp[k0 * 32 + n15];      b0.y = rootp[(k0 + 1) * 32 + n15];
        v2f b1; b1.x = rootp[k0 * 32 + 16 + n15]; b1.y = rootp[(k0 + 1) * 32 + 16 + n15];
        acc0 = __builtin_amdgcn_wmma_f32_16x16x4_f32(false, a, false, b0,
                                                     (short)0, acc0, false, false);
        acc1 = __builtin_amdgcn_wmma_f32_16x16x4_f32(false, a, false, b1,
                                                     (short)0, acc1, false, false);
    }

#pragma unroll
    for (int tile = 0; tile < 2; tile++) {
        v8f acc = tile ? acc1 : acc0;
        int chan = tile * 16 + n15;
        float s1 = 0.0f, s2 = 0.0f;
#pragma unroll
        for (int r = 0; r < 8; r++) {
            int node = rowBase + r + 8 * khalf;
            float val = acc[r] + agg[node * PAD + chan] * invdeg[node];
            if (chan < co) val += bias[chan];
            if (isFinal) {
                if (chan == 0) finalOut[node] = val;
            } else {
                float av = (val > 0.0f) ? val : (expf(val) - 1.0f);   // ELU
                if (chan >= co) av = 0.0f;                            // keep pad zero
                hAct[node * PAD + chan] = av;
                s1 += av; s2 += av * av;
            }
        }
        if (!isFinal && chan < co) {
            atomicAdd(&stats[chan],       s1);
            atomicAdd(&stats[PAD + chan], s2);
        }
    }
}

// ---------------------------------------------------------------------------
// BatchNorm apply (biased variance, matching jnp.var)
// ---------------------------------------------------------------------------
__global__ __launch_bounds__(256) void bn_apply_kernel(
    const float* __restrict__ hAct, const float* __restrict__ stats,
    const float* __restrict__ gamma, const float* __restrict__ beta,
    float* __restrict__ hOut, int co)
{
    int idx = blockIdx.x * blockDim.x + threadIdx.x;
    if (idx >= N_NODES * PAD) return;
    int chan = idx & (PAD - 1);
    float o = 0.0f;
    if (chan < co) {
        float mean = stats[chan]       * (1.0f / N_NODES);
        float ex2  = stats[PAD + chan] * (1.0f / N_NODES);
        float var  = ex2 - mean * mean;
        o = (hAct[idx] - mean) * rsqrtf(var + EPS_BN) * gamma[chan] + beta[chan];
    }
    hOut[idx] = o;
}

// ---------------------------------------------------------------------------
static const int g_chans[12][2] = {{1,8},{8,16},{16,32},{32,32},{32,32},{32,32},
                                   {32,32},{32,32},{32,32},{32,16},{16,8},{8,1}};

static void launch_edge_tile(int L, const float* h, const int* src, const int* dst,
                             const int* sortedE, const float* sortedBas,
                             const int* tileK, const int* tileBase, const int* nTiles,
                             const float* W, float* agg, hipStream_t stream)
{
    dim3 g(MAXTILES / 8), b(256);
    switch (L) {
        case 0:  edge_tile_kernel<1, 8 ><<<g,b,0,stream>>>(h,src,dst,sortedE,sortedBas,tileK,tileBase,nTiles,W,agg); break;
        case 1:  edge_tile_kernel<8, 16><<<g,b,0,stream>>>(h,src,dst,sortedE,sortedBas,tileK,tileBase,nTiles,W,agg); break;
        case 2:  edge_tile_kernel<16,32><<<g,b,0,stream>>>(h,src,dst,sortedE,sortedBas,tileK,tileBase,nTiles,W,agg); break;
        case 9:  edge_tile_kernel<32,16><<<g,b,0,stream>>>(h,src,dst,sortedE,sortedBas,tileK,tileBase,nTiles,W,agg); break;
        case 10: edge_tile_kernel<16,8 ><<<g,b,0,stream>>>(h,src,dst,sortedE,sortedBas,tileK,tileBase,nTiles,W,agg); break;
        case 11: edge_tile_kernel<8, 1 ><<<g,b,0,stream>>>(h,src,dst,sortedE,sortedBas,tileK,tileBase,nTiles,W,agg); break;
        default: edge_tile_kernel<32,32><<<g,b,0,stream>>>(h,src,dst,sortedE,sortedBas,tileK,tileBase,nTiles,W,agg); break;
    }
}

extern "C" void kernel_launch(void* const* d_in, const int* in_sizes, int n_in,
                              void* d_out, int out_size, void* d_ws, size_t ws_size,
                              hipStream_t stream)
{
    const float* x      = (const float*)d_in[0];
    const int*   eix    = (const int*)  d_in[1];
    const int*   src    = eix;
    const int*   dst    = eix + N_EDGES;
    const float* pseudo = (const float*)d_in[2];
    const float *Ws[12], *roots[12], *biases[12], *gammas[11], *betas[11];
    for (int i = 0; i < 12; i++) {
        Ws[i]     = (const float*)d_in[3  + i];
        roots[i]  = (const float*)d_in[15 + i];
        biases[i] = (const float*)d_in[27 + i];
    }
    for (int i = 0; i < 11; i++) {
        gammas[i] = (const float*)d_in[39 + i];
        betas[i]  = (const float*)d_in[50 + i];
    }

    // ---------------- workspace layout ----------------
    float* ws        = (float*)d_ws;
    float* hA        = ws;                           // 8192*32
    float* hB        = hA       + N_NODES * PAD;
    float* hAct      = hB       + N_NODES * PAD;
    float* agg       = hAct     + N_NODES * PAD;
    float* deg       = agg      + N_NODES * PAD;     // 8192
    float* invdeg    = deg      + N_NODES;           // 8192
    float* stats     = invdeg   + N_NODES;           // 64
    float* ebas      = stats    + 64;                // 524288
    float* sortedBas = ebas     + N_ITEMS;           // 774400
    int*   ecorner   = (int*)(sortedBas + MAXPAD);   // 524288
    int*   sortedE   = ecorner  + N_ITEMS;           // 774400
    int*   cnt       = sortedE  + MAXPAD;            // 15625
    int*   tileOffL  = cnt      + NBINS;             // 15625
    int*   padOffL   = tileOffL + NBINS;             // 15625
    int*   fill      = padOffL  + NBINS;             // 15625
    int*   blkNT     = fill     + NBINS;             // 62
    int*   blkPC     = blkNT    + SCAN_BLOCKS;       // 62
    int*   nTiles    = blkPC    + SCAN_BLOCKS;       // 1
    int*   tileK     = nTiles   + 1;                 // 48400
    int*   tileBase  = tileK    + MAXTILES;          // 48400

    // ---------------- one-time precompute (layer-invariant) ----------------
    hipMemsetAsync(deg,       0, N_NODES * sizeof(float), stream);
    hipMemsetAsync(cnt,       0, NBINS * sizeof(int), stream);
    hipMemsetAsync(fill,      0, NBINS * sizeof(int), stream);
    hipMemsetAsync(sortedE,   0, MAXPAD * sizeof(int), stream);
    hipMemsetAsync(sortedBas, 0, MAXPAD * sizeof(float), stream);

    precompute_kernel<<<N_EDGES / 256, 256, 0, stream>>>(pseudo, dst, ebas, ecorner, deg, cnt);
    invdeg_kernel<<<N_NODES / 256, 256, 0, stream>>>(deg, invdeg);
    init_h_kernel<<<(N_NODES * PAD) / 256, 256, 0, stream>>>(x, hA);

    scan_local_kernel<<<SCAN_BLOCKS, 256, 0, stream>>>(cnt, tileOffL, padOffL, blkNT, blkPC);
    scan_blocks_kernel<<<1, 64, 0, stream>>>(blkNT, blkPC, nTiles);
    build_tiles_kernel<<<SCAN_BLOCKS, 256, 0, stream>>>(cnt, tileOffL, padOffL,
                                                        blkNT, blkPC, tileK, tileBase);
    scatter_kernel<<<N_ITEMS / 256, 256, 0, stream>>>(ecorner, ebas, padOffL, fill,
                                                      sortedE, sortedBas);

    // ---------------- 12 spline-conv layers ----------------
    float* cur = hA;
    float* nxt = hB;
    for (int L = 0; L < 12; L++) {
        int ci = g_chans[L][0], co = g_chans[L][1];
        hipMemsetAsync(agg,   0, N_NODES * PAD * sizeof(float), stream);
        hipMemsetAsync(stats, 0, 64 * sizeof(float), stream);

        launch_edge_tile(L, cur, src, dst, sortedE, sortedBas, tileK, tileBase,
                         nTiles, Ws[L], agg, stream);

        int isFinal = (L == 11) ? 1 : 0;
        node_gemm_kernel<<<N_NODES / 128, 256, 0, stream>>>(
            cur, agg, invdeg, roots[L], biases[L], hAct, stats,
            (float*)d_out, ci, co, isFinal);

        if (!isFinal) {
            bn_apply_kernel<<<(N_NODES * PAD) / 256, 256, 0, stream>>>(
                hAct, stats, gammas[L], betas[L], nxt, co);
            float* tmp = cur; cur = nxt; nxt = tmp;
        }
    }
    (void)in_sizes; (void)n_in; (void)out_size; (void)ws_size;
}